// Seq2SeqModel_12498354831414
// MI455X (gfx1250) — compile-verified
//
#include <hip/hip_runtime.h>
#include <hip/hip_bf16.h>

typedef __attribute__((ext_vector_type(2))) float v2f;
typedef __attribute__((ext_vector_type(8))) float v8f;

namespace {

constexpr int Bk = 4096;   // batch
constexpr int Lk = 168;    // encoder sequence length (even: parity carry works)
constexpr int Fk = 8;      // input features
constexpr int Hk = 64;     // hidden size
constexpr int Tk = 24;     // decoder steps
constexpr int BT = 32;     // batch rows per workgroup
constexpr int HS = 66;     // padded LDS row stride (dwords) for h/c state

__device__ __forceinline__ float sigm(float x) {
  return __builtin_amdgcn_rcpf(1.0f + __expf(-x));   // v_rcp_f32, no div fixup
}

__device__ __forceinline__ float tanh_fast(float x) {
  float e = __expf(-2.0f * fabsf(x));
  float t = (1.0f - e) * __builtin_amdgcn_rcpf(1.0f + e);
  return copysignf(t, x);
}

// D = A(16x4) * B(4x16) + C, fp32 WMMA
__device__ __forceinline__ v8f wmma4(v2f a, v2f b, v8f c) {
  return __builtin_amdgcn_wmma_f32_16x16x4_f32(
      /*neg_a=*/false, a, /*neg_b=*/false, b,
      /*c_mod=*/(short)0, c, /*reuse_a=*/false, /*reuse_b=*/false);
}

// Register-resident B-fragments for the 64(k) x 64(4 gates x 16 cols) weight slice owned by this
// wave: [kstep 0..15][gate 0..3], each v2f = 2 VGPRs -> 128 VGPRs.
struct WFrag {
  v2f b[16][4];
};

__device__ __forceinline__ void load_wfrag(WFrag& w, const float* __restrict__ W,
                                           int ncol0, int l15, int lhalf) {
  const int kp = 2 * lhalf;
#pragma unroll
  for (int s = 0; s < 16; ++s) {
    const int k = 4 * s + kp;
#pragma unroll
    for (int g = 0; g < 4; ++g) {
      const int n = g * 64 + ncol0 + l15;
      w.b[s][g] = *(const v2f*)(W + n * Hk + k);
    }
  }
}

// acc[g] += hin(16x64 tile at rows Moff..) @ W^T using register B-fragments.
__device__ __forceinline__ void gemm64_reg(v8f acc[4], const WFrag& w,
                                           const float* hin, int Moff,
                                           int l15, int lhalf) {
  const int kp = 2 * lhalf;
#pragma unroll
  for (int s = 0; s < 16; ++s) {
    const int k = 4 * s + kp;
    v2f a = *(const v2f*)(hin + (Moff + l15) * HS + k);
#pragma unroll
    for (int g = 0; g < 4; ++g) acc[g] = wmma4(a, w.b[s][g], acc[g]);
  }
}

// LSTM cell elementwise update; biases folded here so accumulators can start
// from the inline-0 WMMA SRC2. c state is exclusively owned (no sync needed);
// h is written to the alternate (double) buffer.
__device__ __forceinline__ void cellwise_b(const v8f acc[4], const float bias[4],
                                           float* hw, float* cbuf,
                                           int Moff, int col, int lhalf) {
#pragma unroll
  for (int v = 0; v < 8; ++v) {
    const int row = Moff + v + 8 * lhalf;
    float gi = acc[0][v] + bias[0];
    float gf = acc[1][v] + bias[1];
    float gg = acc[2][v] + bias[2];
    float go = acc[3][v] + bias[3];
    float cc = cbuf[row * HS + col];
    float cn = sigm(gf) * cc + sigm(gi) * tanh_fast(gg);
    float hn = sigm(go) * tanh_fast(cn);
    cbuf[row * HS + col] = cn;
    hw[row * HS + col] = hn;
  }
}

__global__ __launch_bounds__(256) void lstm_seq2seq_kernel(
    const float* __restrict__ x,
    const float* __restrict__ eWih0, const float* __restrict__ eWhh0, const float* __restrict__ eb0,
    const float* __restrict__ eWih1, const float* __restrict__ eWhh1, const float* __restrict__ eb1,
    const float* __restrict__ dWih0, const float* __restrict__ dWhh0, const float* __restrict__ db0,
    const float* __restrict__ dWih1, const float* __restrict__ dWhh1, const float* __restrict__ db1,
    const float* __restrict__ Wout, const float* __restrict__ bout,
    float* __restrict__ out) {
  // double-buffered h (removes read/write hazard -> 2 barriers per timestep)
  __shared__ float h0a[BT * HS], h0b[BT * HS];
  __shared__ float h1a[BT * HS], h1b[BT * HS];
  __shared__ float c0[BT * HS], c1[BT * HS];
  __shared__ float ybuf[BT];
  __shared__ float pbuf[BT * 8];

  const int tid = threadIdx.x;
  const int lane = tid & 31;
  const int wave = tid >> 5;
  const int l15 = lane & 15;
  const int lhalf = lane >> 4;
  const int kp = 2 * lhalf;
  const int Moff = (wave & 1) * 16;      // batch-row tile within the 32-row block
  const int ncol0 = (wave >> 1) * 16;    // hidden-column tile (per-gate)
  const int col = ncol0 + l15;
  const int b0 = blockIdx.x * BT;

  for (int i = tid; i < BT * HS; i += 256) {
    h0a[i] = 0.0f; h0b[i] = 0.0f; h1a[i] = 0.0f; h1b[i] = 0.0f;
    c0[i] = 0.0f;  c1[i] = 0.0f;
  }
  if (tid < BT) ybuf[tid] = 0.0f;
  __syncthreads();

  float bias0[4], bias1[4];
#pragma unroll
  for (int g = 0; g < 4; ++g) {
    const int n = g * 64 + col;
    bias0[g] = eb0[n];
    bias1[g] = eb1[n];
  }

  // encoder layer-0 input-projection fragments (K=8): 16 VGPRs
  v2f wx[2][4];
#pragma unroll
  for (int s = 0; s < 2; ++s)
#pragma unroll
    for (int g = 0; g < 4; ++g)
      wx[s][g] = *(const v2f*)(eWih0 + (g * 64 + col) * Fk + 4 * s + kp);

  // Recurrent weight fragments resident in registers (3 x 128 = 384 VGPRs).
  WFrag wA, wB, wC;
  load_wfrag(wA, eWhh0, ncol0, l15, lhalf);
  load_wfrag(wB, eWih1, ncol0, l15, lhalf);
  load_wfrag(wC, eWhh1, ncol0, l15, lhalf);

  // software-pipelined x fragments (loaded one timestep ahead)
  const float* xbase = x + (size_t)(b0 + Moff + l15) * Lk * Fk;
  v2f xa0 = *(const v2f*)(xbase + kp);
  v2f xa1 = *(const v2f*)(xbase + 4 + kp);

  // ----------------- encoder: 168 steps, 2 fused layers -----------------
  for (int t = 0; t < Lk; ++t) {
    const float* h0r = (t & 1) ? h0b : h0a;
    float*       h0w = (t & 1) ? h0a : h0b;
    const float* h1r = (t & 1) ? h1b : h1a;
    float*       h1w = (t & 1) ? h1a : h1b;

    v2f xc0 = xa0, xc1 = xa1;
    if (t + 1 < Lk) {  // prefetch next timestep's input
      xa0 = *(const v2f*)(xbase + (t + 1) * Fk + kp);
      xa1 = *(const v2f*)(xbase + (t + 1) * Fk + 4 + kp);
    }

    // layer 0: gates = x_t @ Wih0^T + h0 @ Whh0^T  (bias folded in cellwise)
    v8f acc[4] = {};  // inline-0 SRC2 on first WMMA, no broadcast movs
#pragma unroll
    for (int g = 0; g < 4; ++g) acc[g] = wmma4(xc0, wx[0][g], acc[g]);
#pragma unroll
    for (int g = 0; g < 4; ++g) acc[g] = wmma4(xc1, wx[1][g], acc[g]);
    gemm64_reg(acc, wA, h0r, Moff, l15, lhalf);
    cellwise_b(acc, bias0, h0w, c0, Moff, col, lhalf);
    __syncthreads();              // fresh h0 visible to all waves

    // layer 1: gates = h0_new @ Wih1^T + h1 @ Whh1^T
    v8f acc2[4] = {};
    gemm64_reg(acc2, wB, h0w, Moff, l15, lhalf);
    gemm64_reg(acc2, wC, h1r, Moff, l15, lhalf);
    cellwise_b(acc2, bias1, h1w, c1, Moff, col, lhalf);
    __syncthreads();              // fresh h1 / parity flip
  }

  // ----------------- decoder: 24 autoregressive steps -----------------
  // Lk is even, so decoder t&1 parity continues seamlessly from encoder finals.
  float dbias0[4], dbias1[4], wih0d[4];
#pragma unroll
  for (int g = 0; g < 4; ++g) {
    const int n = g * 64 + col;
    dbias0[g] = db0[n];
    dbias1[g] = db1[n];
    wih0d[g] = dWih0[n];
  }
  load_wfrag(wA, dWhh0, ncol0, l15, lhalf);   // reuse the same 384 VGPRs
  load_wfrag(wB, dWih1, ncol0, l15, lhalf);
  load_wfrag(wC, dWhh1, ncol0, l15, lhalf);

  float wout_r[8];
  {
    const int seg = tid >> 5;
#pragma unroll
    for (int q = 0; q < 8; ++q) wout_r[q] = Wout[seg * 8 + q];
  }
  const float bout0 = bout[0];

  for (int t = 0; t < Tk; ++t) {
    const float* h0r = (t & 1) ? h0b : h0a;
    float*       h0w = (t & 1) ? h0a : h0b;
    const float* h1r = (t & 1) ? h1b : h1a;
    float*       h1w = (t & 1) ? h1a : h1b;

    // layer 0: scalar input y per row (K=1 folded into accumulator init)
    v8f acc[4];
#pragma unroll
    for (int g = 0; g < 4; ++g)
#pragma unroll
      for (int v = 0; v < 8; ++v) {
        const int row = Moff + v + 8 * lhalf;
        acc[g][v] = ybuf[row] * wih0d[g];
      }
    gemm64_reg(acc, wA, h0r, Moff, l15, lhalf);
    cellwise_b(acc, dbias0, h0w, c0, Moff, col, lhalf);
    __syncthreads();

    // layer 1
    v8f acc2[4] = {};
    gemm64_reg(acc2, wB, h0w, Moff, l15, lhalf);
    gemm64_reg(acc2, wC, h1r, Moff, l15, lhalf);
    cellwise_b(acc2, dbias1, h1w, c1, Moff, col, lhalf);
    __syncthreads();

    // projection y = h1 @ Wout^T + bout  (two-stage LDS reduction)
    {
      const int r = tid & 31;
      const int seg = tid >> 5;
      float sum = 0.0f;
#pragma unroll
      for (int q = 0; q < 8; ++q) sum += h1w[r * HS + seg * 8 + q] * wout_r[q];
      pbuf[r * 8 + seg] = sum;
    }
    __syncthreads();
    if (tid < BT) {
      float y = bout0;
#pragma unroll
      for (int s = 0; s < 8; ++s) y += pbuf[tid * 8 + s];
      ybuf[tid] = y;
      out[(size_t)(b0 + tid) * Tk + t] = y;
    }
    __syncthreads();
  }
}

}  // namespace

extern "C" void kernel_launch(void* const* d_in, const int* in_sizes, int n_in,
                              void* d_out, int out_size, void* d_ws, size_t ws_size,
                              hipStream_t stream) {
  (void)in_sizes; (void)n_in; (void)out_size; (void)d_ws; (void)ws_size;
  const float* x     = (const float*)d_in[0];
  // d_in[1] = target_len (compile-time constant Tk = 24)
  const float* eWih0 = (const float*)d_in[2];
  const float* eWhh0 = (const float*)d_in[3];
  const float* eb0   = (const float*)d_in[4];
  const float* eWih1 = (const float*)d_in[5];
  const float* eWhh1 = (const float*)d_in[6];
  const float* eb1   = (const float*)d_in[7];
  const float* dWih0 = (const float*)d_in[8];
  const float* dWhh0 = (const float*)d_in[9];
  const float* db0   = (const float*)d_in[10];
  const float* dWih1 = (const float*)d_in[11];
  const float* dWhh1 = (const float*)d_in[12];
  const float* db1   = (const float*)d_in[13];
  const float* Wout  = (const float*)d_in[14];
  const float* bout  = (const float*)d_in[15];
  float* out = (float*)d_out;

  dim3 grid(Bk / BT);   // 128 workgroups, each owns 32 batch rows end-to-end
  dim3 block(256);      // 8 wave32s
  lstm_seq2seq_kernel<<<grid, block, 0, stream>>>(
      x, eWih0, eWhh0, eb0, eWih1, eWhh1, eb1,
      dWih0, dWhh0, db0, dWih1, dWhh1, db1, Wout, bout, out);
}